// GCNReg_1563368096536
// MI455X (gfx1250) — compile-verified
//
#include <hip/hip_runtime.h>

typedef float v2f __attribute__((ext_vector_type(2)));
typedef float v8f __attribute__((ext_vector_type(8)));
typedef int   vi4 __attribute__((vector_size(16)));   // param type of async-LDS builtin

#define FDIM 128   // input features
#define HDIM 64    // hidden features

#if defined(__gfx1250__) && __has_builtin(__builtin_amdgcn_global_load_async_to_lds_b128)
#define HAVE_ASYNC_LDS 1
#else
#define HAVE_ASYNC_LDS 0
#endif

// ---------------- degree / norm ----------------
__global__ void k_init_deg(unsigned* __restrict__ deg, int N) {
  int i = blockIdx.x * blockDim.x + threadIdx.x;
  if (i < N) deg[i] = 1u;   // self-loop contributes 1
}

__global__ void k_deg(const long long* __restrict__ col, unsigned* __restrict__ deg, int E) {
  int e = blockIdx.x * blockDim.x + threadIdx.x;
  if (e < E) atomicAdd(&deg[(int)col[e]], 1u);
}

__global__ void k_dinv(const unsigned* __restrict__ deg, float* __restrict__ dinv, int N) {
  int i = blockIdx.x * blockDim.x + threadIdx.x;
  if (i < N) dinv[i] = rsqrtf((float)deg[i]);   // deg >= 1 always
}

// ---------------- layer-1 GEMM: h = x @ W1 via f32 WMMA ----------------
// Block: 128 threads = 4 waves, each wave computes a 16x64 output tile.
// A-frag (16x4 f32): lane L (L<16): M=L, K=k0+{0,1}; lane L+16: M=L, K=k0+{2,3}.
// B-frag (4x16 f32): lane L (L<16): N=L, K=k0+{0,1}; lane L+16: N=L, K=k0+{2,3}.
// C/D (16x16 f32): vgpr r, lanes<16 -> M=r, lanes>=16 -> M=8+r, N=lane&15.
__global__ __launch_bounds__(128) void k_gemm_wmma(
    const float* __restrict__ x, const float* __restrict__ W,
    float* __restrict__ h, int N) {
  __shared__ float xs[64][132];   // 64 rows x 128 cols, stride 132 (bank spread + 16B align)

  const int t = threadIdx.x;
  const long rowBase = (long)blockIdx.x * 64;

  // Stage the 64x128 x-tile into LDS. On gfx1250, use the async copy engine
  // (ASYNCcnt-tracked, writes LDS directly without touching VGPRs).
#pragma unroll
  for (int i = 0; i < 16; ++i) {
    int lin = i * 512 + t * 4;         // 0..8191
    int r = lin >> 7;
    int c = lin & 127;
    long gr = rowBase + r;
    float* ldst = &xs[r][c];
#if HAVE_ASYNC_LDS
    if (gr < N) {
      __builtin_amdgcn_global_load_async_to_lds_b128(
          (vi4*)(x + gr * FDIM + c), (vi4*)ldst, /*offset=*/0, /*cpol=*/0);
    } else {
      *(float4*)ldst = float4{0.f, 0.f, 0.f, 0.f};   // zero-pad OOB rows
    }
#else
    float4 v = {0.f, 0.f, 0.f, 0.f};
    if (gr < N) v = *(const float4*)(x + gr * FDIM + c);
    *(float4*)ldst = v;
#endif
  }
#if HAVE_ASYNC_LDS
#if __has_builtin(__builtin_amdgcn_s_wait_asynccnt)
  __builtin_amdgcn_s_wait_asynccnt(0);
#else
  asm volatile("s_wait_asynccnt 0x0" ::: "memory");
#endif
#endif
  __syncthreads();

  const int lane = t & 31;
  const int wv = t >> 5;             // wave id: rows wv*16 .. wv*16+15
  const int g = lane >> 4;           // half-wave selector (K offset 0 or 2)
  const int n = lane & 15;           // A row / B col within tile
  const int m = wv * 16 + n;         // LDS row for A fragment

  v8f acc0 = {}, acc1 = {}, acc2 = {}, acc3 = {};

#pragma unroll 4
  for (int k0 = 0; k0 < FDIM; k0 += 4) {
    const int kb = k0 + 2 * g;
    v2f a;
    a.x = xs[m][kb];
    a.y = xs[m][kb + 1];

    const float* w0 = W + (long)kb * HDIM + n;        // row kb of W1 (L1-resident, 32KB)
    const float* w1 = w0 + HDIM;                      // row kb+1 of W1
    v2f b0; b0.x = w0[0];  b0.y = w1[0];
    v2f b1; b1.x = w0[16]; b1.y = w1[16];
    v2f b2; b2.x = w0[32]; b2.y = w1[32];
    v2f b3; b3.x = w0[48]; b3.y = w1[48];

    acc0 = __builtin_amdgcn_wmma_f32_16x16x4_f32(false, a, false, b0, (short)0, acc0, false, false);
    acc1 = __builtin_amdgcn_wmma_f32_16x16x4_f32(false, a, false, b1, (short)0, acc1, false, false);
    acc2 = __builtin_amdgcn_wmma_f32_16x16x4_f32(false, a, false, b2, (short)0, acc2, false, false);
    acc3 = __builtin_amdgcn_wmma_f32_16x16x4_f32(false, a, false, b3, (short)0, acc3, false, false);
  }

  // Store D: row = base + wv*16 + g*8 + r ; col = nt*16 + n
  const long row0 = rowBase + wv * 16 + g * 8;
#pragma unroll
  for (int r = 0; r < 8; ++r) {
    long gr = row0 + r;
    if (gr < N) {
      float* dst = h + gr * HDIM + n;
      dst[0]  = acc0[r];
      dst[16] = acc1[r];
      dst[32] = acc2[r];
      dst[48] = acc3[r];
    }
  }
}

// ---------------- self-loop init: agg[i] = h[i] * dinv[i]^2 (no atomics) ----------------
__global__ void k_selfinit(const float* __restrict__ h, const float* __restrict__ dinv,
                           float* __restrict__ agg, long total) {
  long i = (long)blockIdx.x * blockDim.x + threadIdx.x;
  if (i < total) {
    float d = dinv[i >> 6];
    agg[i] = h[i] * d * d;
  }
}

// ---------------- layer-1 edge aggregation: wave per edge, lane per float2 ----------------
__global__ __launch_bounds__(256) void k_agg1(
    const long long* __restrict__ rowA, const long long* __restrict__ colA,
    const float* __restrict__ dinv, const float* __restrict__ h,
    float* __restrict__ agg, int E) {
  int e = blockIdx.x * 8 + (threadIdx.x >> 5);
  int lane = threadIdx.x & 31;
  if (e >= E) return;
  int r = (int)rowA[e];
  int c = (int)colA[e];
  __builtin_prefetch(h + (long)r * HDIM + 2 * lane, 0, 0);  // global_prefetch_b8
  float w = dinv[r] * dinv[c];
  float2 hv = *(const float2*)(h + (long)r * HDIM + 2 * lane);
  float* dst = agg + (long)c * HDIM + 2 * lane;
  unsafeAtomicAdd(dst,     hv.x * w);   // global_atomic_add_f32, L2-resident target
  unsafeAtomicAdd(dst + 1, hv.y * w);
}

// ---------------- relu + bias + dot(W2) + out init (wave per node) ----------------
__global__ __launch_bounds__(256) void k_relu_dot(
    const float* __restrict__ agg, const float* __restrict__ b1,
    const float* __restrict__ W2, const float* __restrict__ b2,
    const float* __restrict__ dinv, float* __restrict__ z,
    float* __restrict__ out, int N) {
  int lane = threadIdx.x & 31;
  long i = (long)blockIdx.x * 8 + (threadIdx.x >> 5);
  if (i >= N) return;
  float2 a = *(const float2*)(agg + i * HDIM + 2 * lane);
  float p = fmaxf(a.x + b1[2 * lane], 0.f) * W2[2 * lane]
          + fmaxf(a.y + b1[2 * lane + 1], 0.f) * W2[2 * lane + 1];
#pragma unroll
  for (int off = 16; off > 0; off >>= 1) p += __shfl_xor(p, off, 32);
  if (lane == 0) {
    z[i] = p;
    float d = dinv[i];
    out[i] = b2[0] + p * d * d;   // b2 + self-loop term; fully overwrites poisoned d_out
  }
}

// ---------------- layer-2 edge aggregation: 1 scalar atomic per edge ----------------
__global__ void k_agg2(const long long* __restrict__ rowA, const long long* __restrict__ colA,
                       const float* __restrict__ dinv, const float* __restrict__ z,
                       float* __restrict__ out, int E) {
  int e = blockIdx.x * blockDim.x + threadIdx.x;
  if (e >= E) return;
  int r = (int)rowA[e];
  int c = (int)colA[e];
  unsafeAtomicAdd(out + c, z[r] * dinv[r] * dinv[c]);
}

extern "C" void kernel_launch(void* const* d_in, const int* in_sizes, int n_in,
                              void* d_out, int out_size, void* d_ws, size_t ws_size,
                              hipStream_t stream) {
  const float* x      = (const float*)d_in[0];
  const long long* ei = (const long long*)d_in[1];   // int64 edge_index [2,E]
  const float* W1     = (const float*)d_in[2];
  const float* b1     = (const float*)d_in[3];
  const float* W2     = (const float*)d_in[4];
  const float* b2     = (const float*)d_in[5];
  float* out = (float*)d_out;

  const int N = in_sizes[0] / FDIM;
  const int E = in_sizes[1] / 2;
  const long long* rowA = ei;
  const long long* colA = ei + E;

  // Workspace carve-out (≈52.5 MB): deg, dinv, z, h1, agg1
  char* ws = (char*)d_ws;
  size_t off = 0;
  auto take = [&](size_t bytes) {
    char* p = ws + off;
    off = (off + bytes + 255) & ~(size_t)255;
    return p;
  };
  unsigned* deg = (unsigned*)take((size_t)N * 4);
  float* dinv   = (float*)take((size_t)N * 4);
  float* z      = (float*)take((size_t)N * 4);
  float* h1     = (float*)take((size_t)N * HDIM * 4);
  float* agg1   = (float*)take((size_t)N * HDIM * 4);
  (void)ws_size; (void)n_in; (void)out_size;

  k_init_deg<<<(N + 255) / 256, 256, 0, stream>>>(deg, N);
  k_deg<<<(E + 255) / 256, 256, 0, stream>>>(colA, deg, E);
  k_dinv<<<(N + 255) / 256, 256, 0, stream>>>(deg, dinv, N);
  k_gemm_wmma<<<(N + 63) / 64, 128, 0, stream>>>(x, W1, h1, N);
  long total = (long)N * HDIM;
  k_selfinit<<<(unsigned)((total + 255) / 256), 256, 0, stream>>>(h1, dinv, agg1, total);
  k_agg1<<<(E + 7) / 8, 256, 0, stream>>>(rowA, colA, dinv, h1, agg1, E);
  k_relu_dot<<<(N + 7) / 8, 256, 0, stream>>>(agg1, b1, W2, b2, dinv, z, out, N);
  k_agg2<<<(E + 255) / 256, 256, 0, stream>>>(rowA, colA, dinv, z, out, E);
}